// GraphDecoder_47055661695544
// MI455X (gfx1250) — compile-verified
//
#include <hip/hip_runtime.h>

typedef float v2f __attribute__((ext_vector_type(2)));
typedef float v8f __attribute__((ext_vector_type(8)));

#define N_NODES 1024
#define EMB_D   128
#define HID_H   256

// ---------------------------------------------------------------------------
// Async global->LDS copy of 16 bytes (CDNA5 GLOBAL_LOAD_ASYNC_TO_LDS_B128).
// ldsOff: wave-relative LDS byte offset (low 32 bits of generic shared ptr).
// gaddr : 64-bit global address. Tracked by ASYNCcnt.
// ---------------------------------------------------------------------------
__device__ __forceinline__ void async_copy_b128(uint32_t ldsOff, const void* gptr) {
  asm volatile("global_load_async_to_lds_b128 %0, %1, off"
               :: "v"(ldsOff), "v"((uint64_t)(uintptr_t)gptr)
               : "memory");
}
__device__ __forceinline__ void async_wait0() {
  asm volatile("s_wait_asynccnt 0x0" ::: "memory");
}
__device__ __forceinline__ uint32_t lds_off(const void* sptr) {
  // generic pointers to LDS carry the LDS byte offset in their low 32 bits
  return (uint32_t)(uintptr_t)sptr;
}

// ---------------------------------------------------------------------------
// Wave-level f32 GEMM using V_WMMA_F32_16X16X4_F32 (full precision, wave32).
// One wave (32 lanes) computes one 16x16 tile of C = A[MxK] @ B[KxN] (+bias,+relu).
// A row-major [M,K], B row-major [K,N], C row-major [M,N]. M,N %16==0, K %8==0.
//
// Fragment layout (CDNA5 ISA 7.12.2):
//   A 16x4 f32: lane%16 = M row, (lane/16)*2 = K base, 2 VGPRs = K, K+1
//   B 4x16 f32: lane%16 = N col, (lane/16)*2 = K base, 2 VGPRs = K, K+1
//   C/D 16x16 f32: VGPR r -> (M=r, N=lane) lanes 0-15 ; (M=r+8, N=lane-16)
// ---------------------------------------------------------------------------
template<bool RELU>
__global__ void gemm_wmma_f32(const float* __restrict__ A,
                              const float* __restrict__ B,
                              const float* __restrict__ bias,   // [N] or nullptr
                              float* __restrict__ C,
                              int M, int N, int K)
{
  const int lane   = threadIdx.x & 31;
  const int tile_n = blockIdx.x * blockDim.y + threadIdx.y;
  const int tile_m = blockIdx.y;
  const int row0   = tile_m << 4;
  const int col0   = tile_n << 4;
  const int mn     = lane & 15;          // A: row within tile, B: col within tile
  const int kg     = (lane >> 4) << 1;   // k sub-offset {0,2}

  v8f acc0 = {};
  v8f acc1 = {};
  const float* aBase = A + (size_t)(row0 + mn) * K + kg;   // contiguous pair in K
  const float* bBase = B + (size_t)kg * N + (col0 + mn);   // stride-N pair in K

  for (int kk = 0; kk < K; kk += 8) {
    // two interleaved K=4 steps -> two accumulators hide the WMMA RAW hazard
    v2f a0 = *(const v2f*)(aBase + kk);
    v2f b0; b0.x = bBase[(size_t)kk * N];       b0.y = bBase[(size_t)(kk + 1) * N];
    acc0 = __builtin_amdgcn_wmma_f32_16x16x4_f32(false, a0, false, b0,
                                                 (short)0, acc0, false, false);
    v2f a1 = *(const v2f*)(aBase + kk + 4);
    v2f b1; b1.x = bBase[(size_t)(kk + 4) * N]; b1.y = bBase[(size_t)(kk + 5) * N];
    acc1 = __builtin_amdgcn_wmma_f32_16x16x4_f32(false, a1, false, b1,
                                                 (short)0, acc1, false, false);
  }
  v8f acc = acc0 + acc1;

  const int n     = col0 + mn;
  const int mbase = row0 + ((lane >> 4) << 3);
  const float bv  = bias ? bias[n] : 0.0f;
  float* cp = C + (size_t)mbase * N + n;
#pragma unroll
  for (int r = 0; r < 8; ++r) {
    float v = acc[r] + bv;
    if (RELU) v = fmaxf(v, 0.0f);
    cp[(size_t)r * N] = v;
  }
}

// ---------------------------------------------------------------------------
// Edge kernel: adj[i,j] = sigmoid( sum_h relu(U[i,h]+V[j,h]) * w[h] + be2 ),
// diagonal forced to 0.  16x16 output tile per block.  U/V row panels and w
// are staged into LDS with the CDNA5 async global->LDS engine (no VGPR
// round-trip), then each thread accumulates one (i,j) via float4 LDS reads.
// ---------------------------------------------------------------------------
__global__ void edge_kernel(const float* __restrict__ U,   // [N,H] = hi + be1
                            const float* __restrict__ V,   // [N,H] = hj
                            const float* __restrict__ w,   // [H]   = We2[:,0]
                            const float* __restrict__ be2, // [1]
                            float* __restrict__ adj)       // [N,N]
{
  __shared__ float su[16][HID_H];   // 16 KB
  __shared__ float sv[16][HID_H];   // 16 KB
  __shared__ float sw[HID_H];       //  1 KB

  const int i0 = blockIdx.y << 4;
  const int j0 = blockIdx.x << 4;
  const int t  = threadIdx.y * 16 + threadIdx.x;   // 0..255

  // async stage: 16 rows x 64 float4 from each of U and V, plus w
  for (int idx = t; idx < 16 * 64; idx += 256) {
    const int r = idx >> 6, c4 = idx & 63;
    async_copy_b128(lds_off(&su[r][c4 * 4]),
                    U + (size_t)(i0 + r) * HID_H + c4 * 4);
    async_copy_b128(lds_off(&sv[r][c4 * 4]),
                    V + (size_t)(j0 + r) * HID_H + c4 * 4);
  }
  if (t < 64) async_copy_b128(lds_off(&sw[t * 4]), w + t * 4);

  async_wait0();      // our async writes to LDS are complete
  __syncthreads();    // everyone's are visible

  const int ti = threadIdx.y, tj = threadIdx.x;
  const float4* uq = (const float4*)su[ti];
  const float4* vq = (const float4*)sv[tj];
  const float4* wq = (const float4*)sw;

  float acc = 0.0f;
#pragma unroll 4
  for (int h = 0; h < HID_H / 4; ++h) {
    const float4 a = uq[h], b = vq[h], ww = wq[h];
    acc += fmaxf(a.x + b.x, 0.0f) * ww.x;
    acc += fmaxf(a.y + b.y, 0.0f) * ww.y;
    acc += fmaxf(a.z + b.z, 0.0f) * ww.z;
    acc += fmaxf(a.w + b.w, 0.0f) * ww.w;
  }

  const int i = i0 + ti, j = j0 + tj;
  const float logit = acc + be2[0];
  float s = 1.0f / (1.0f + __expf(-logit));
  if (i == j) s = 0.0f;
  adj[(size_t)i * N_NODES + j] = s;
}

// ---------------------------------------------------------------------------
extern "C" void kernel_launch(void* const* d_in, const int* in_sizes, int n_in,
                              void* d_out, int out_size, void* d_ws, size_t ws_size,
                              hipStream_t stream)
{
  (void)in_sizes; (void)n_in; (void)out_size; (void)ws_size;

  const float* E   = (const float*)d_in[0];   // [N,D]
  const float* W1  = (const float*)d_in[1];   // [D,H]
  const float* b1  = (const float*)d_in[2];   // [H]
  const float* W2  = (const float*)d_in[3];   // [H,D]
  const float* b2  = (const float*)d_in[4];   // [D]
  const float* We1 = (const float*)d_in[5];   // [2D,H]
  const float* be1 = (const float*)d_in[6];   // [H]
  const float* We2 = (const float*)d_in[7];   // [H,1] -> [H]
  const float* be2 = (const float*)d_in[8];   // [1]

  float* adj = (float*)d_out;                                   // [N,N]
  float* nf  = (float*)d_out + (size_t)N_NODES * N_NODES;       // [N,D]

  float* H1 = (float*)d_ws;                                     // [N,H]
  float* U  = H1 + (size_t)N_NODES * HID_H;                     // [N,H]
  float* V  = U  + (size_t)N_NODES * HID_H;                     // [N,H]

  const dim3 blk(32, 4);   // 4 waves, 4 adjacent 16-col tiles per block

  // H1 = relu(E @ W1 + b1)
  gemm_wmma_f32<true><<<dim3(HID_H / 64, N_NODES / 16), blk, 0, stream>>>(
      E, W1, b1, H1, N_NODES, HID_H, EMB_D);

  // node_features = H1 @ W2 + b2
  gemm_wmma_f32<false><<<dim3(EMB_D / 64, N_NODES / 16), blk, 0, stream>>>(
      H1, W2, b2, nf, N_NODES, EMB_D, HID_H);

  // U = E @ We1[:D] + be1   (bias folded so edge kernel skips it)
  gemm_wmma_f32<false><<<dim3(HID_H / 64, N_NODES / 16), blk, 0, stream>>>(
      E, We1, be1, U, N_NODES, HID_H, EMB_D);

  // V = E @ We1[D:]
  gemm_wmma_f32<false><<<dim3(HID_H / 64, N_NODES / 16), blk, 0, stream>>>(
      E, We1 + (size_t)EMB_D * HID_H, nullptr, V, N_NODES, HID_H, EMB_D);

  // adj = sigmoid(edge scores), zero diagonal
  edge_kernel<<<dim3(N_NODES / 16, N_NODES / 16), dim3(16, 16), 0, stream>>>(
      U, V, We2, be2, adj);
}